// MMPrompt_ContextAware_45698452030128
// MI455X (gfx1250) — compile-verified
//
#include <hip/hip_runtime.h>
#include <hip/hip_bf16.h>

// ---------------------------------------------------------------------------
// Problem constants (from the reference)
// ---------------------------------------------------------------------------
#define H      512
#define HALF   256
#define NHEAD  8
#define NLAYER 12
#define NBLOCK 2
#define NENT   50000
#define NREL   12
#define NBASES 8
#define NEDGE  500000
#define TXTD   768
#define BATCH  32
#define ELEN   50
#define SLEN   128
#define NROW   (BATCH * ELEN)          // 1600
#define KCAT   ((NREL + 1) * HALF)     // 3328
#define PPN    (NLAYER * NBLOCK * H)   // 12288

typedef float v2f __attribute__((ext_vector_type(2)));
typedef float v8f __attribute__((ext_vector_type(8)));

// ---------------------------------------------------------------------------
// Utility fills
// ---------------------------------------------------------------------------
__global__ void fill_f32(float* p, float v, int n) {
    int i = blockIdx.x * blockDim.x + threadIdx.x;
    if (i < n) p[i] = v;
}
__global__ void fill_i32(int* p, int v, int n) {
    int i = blockIdx.x * blockDim.x + threadIdx.x;
    if (i < n) p[i] = v;
}

// ---------------------------------------------------------------------------
// Compact unique-entity slot assignment: slot_map[ent] in [0,U), -1 otherwise
// ---------------------------------------------------------------------------
__global__ void assign_slots(const int* __restrict__ eids, int* __restrict__ slot_map,
                             int* __restrict__ counter, int* __restrict__ node_of_slot) {
    int i = blockIdx.x * blockDim.x + threadIdx.x;
    if (i >= NROW) return;
    int id = eids[i];
    int old = atomicCAS(&slot_map[id], -1, -2);
    if (old == -1) {
        int s = atomicAdd(counter, 1);
        node_of_slot[s] = id;
        __threadfence();
        slot_map[id] = s;
    }
}

// ---------------------------------------------------------------------------
// Wcat[3328,256]: rows [r*256+i] = sum_b comp[r,b]*basis[b,i,o]; rows [3072+i] = root[i,o]
// ---------------------------------------------------------------------------
__global__ void build_wcat(const float* __restrict__ basis, const float* __restrict__ comp,
                           const float* __restrict__ root, float* __restrict__ wcat) {
    int idx = blockIdx.x * blockDim.x + threadIdx.x;
    if (idx >= KCAT * HALF) return;
    int row = idx >> 8;          // /256
    int o   = idx & 255;
    if (row < NREL * HALF) {
        int r = row >> 8;
        int i = row & 255;
        float acc = 0.f;
#pragma unroll
        for (int b = 0; b < NBASES; ++b)
            acc += comp[r * NBASES + b] * basis[((size_t)b * HALF + i) * HALF + o];
        wcat[idx] = acc;
    } else {
        wcat[idx] = root[(size_t)(row - NREL * HALF) * HALF + o];
    }
}

// ---------------------------------------------------------------------------
// Edge scan: one wave per edge; scatter x[src] into compact (slot, rel) buckets
// ---------------------------------------------------------------------------
__global__ void edge_scatter(const int* __restrict__ edge_index, const int* __restrict__ edge_type,
                             const int* __restrict__ slot_map, const float* __restrict__ x,
                             float* __restrict__ agg, float* __restrict__ cnt) {
    int wave = blockIdx.x * (blockDim.x >> 5) + (threadIdx.x >> 5);
    if (wave >= NEDGE) return;
    int dst = edge_index[NEDGE + wave];
    int s = slot_map[dst];
    if (s < 0) return;                                   // wave-uniform branch
    int lane = threadIdx.x & 31;
    int rel = edge_type[wave];
    int src = edge_index[wave];
    const float* xs = x + (size_t)src * HALF;
    float* ag = agg + (size_t)(s * NREL + rel) * HALF;
#pragma unroll
    for (int j = 0; j < HALF / 32; ++j) {
        int k = lane + 32 * j;
        atomicAdd(&ag[k], xs[k]);
    }
    if (lane == 0) atomicAdd(&cnt[s * NREL + rel], 1.0f);
}

// ---------------------------------------------------------------------------
// Acat[1600,3328]: [mean(s,r,i) | x[node_of_slot[s], i]]
// ---------------------------------------------------------------------------
__global__ void build_acat(const float* __restrict__ agg, const float* __restrict__ cnt,
                           const int* __restrict__ node_of_slot, const float* __restrict__ x,
                           float* __restrict__ acat) {
    int idx = blockIdx.x * blockDim.x + threadIdx.x;
    if (idx >= NROW * KCAT) return;
    int s = idx / KCAT;
    int c = idx - s * KCAT;
    if (c < NREL * HALF) {
        int r = c >> 8;
        int i = c & 255;
        float cn = cnt[s * NREL + r];
        float d = cn > 1.f ? cn : 1.f;
        acat[idx] = agg[(size_t)(s * NREL + r) * HALF + i] / d;
    } else {
        acat[idx] = x[(size_t)node_of_slot[s] * HALF + (c - NREL * HALF)];
    }
}

// ---------------------------------------------------------------------------
// Row gather: out[i,:] = src[row(i),:]; row = slot_map[eids[i]] or eids[i]
// ---------------------------------------------------------------------------
__global__ void gather_rows(const float* __restrict__ src, const int* __restrict__ eids,
                            const int* __restrict__ slot_map, float* __restrict__ out, int C) {
    int idx = blockIdx.x * blockDim.x + threadIdx.x;
    if (idx >= NROW * C) return;
    int i = idx / C;
    int j = idx - i * C;
    int id = eids[i];
    int row = slot_map ? slot_map[id] : id;
    out[idx] = src[(size_t)row * C + j];
}

// ---------------------------------------------------------------------------
// q input: mean over SLEN tokens
// ---------------------------------------------------------------------------
__global__ void qmean(const float* __restrict__ tok, float* __restrict__ qin) {
    int idx = blockIdx.x * blockDim.x + threadIdx.x;
    if (idx >= BATCH * H) return;
    int b = idx / H;
    int j = idx - b * H;
    float s = 0.f;
    for (int l = 0; l < SLEN; ++l) s += tok[((size_t)b * SLEN + l) * H + j];
    qin[idx] = s * (1.0f / SLEN);
}

// ---------------------------------------------------------------------------
// Context-gated fusion: one wave per (b,l); dot(q, k-proj) x3 -> softmax -> blend
// ---------------------------------------------------------------------------
__global__ void fuse_kernel(const float* __restrict__ q,
                            const float* __restrict__ kg, const float* __restrict__ kk,
                            const float* __restrict__ im, const float* __restrict__ ik,
                            const float* __restrict__ tx, const float* __restrict__ tk,
                            float* __restrict__ fused) {
    int wave = blockIdx.x * (blockDim.x >> 5) + (threadIdx.x >> 5);
    if (wave >= NROW) return;
    int lane = threadIdx.x & 31;
    const float* qb = q + (size_t)(wave / ELEN) * H;
    size_t base = (size_t)wave * H;
    float s0 = 0.f, s1 = 0.f, s2 = 0.f;
    for (int j = lane; j < H; j += 32) {
        float qv = qb[j];
        s0 += qv * kk[base + j];
        s1 += qv * ik[base + j];
        s2 += qv * tk[base + j];
    }
#pragma unroll
    for (int off = 16; off > 0; off >>= 1) {
        s0 += __shfl_xor(s0, off, 32);
        s1 += __shfl_xor(s1, off, 32);
        s2 += __shfl_xor(s2, off, 32);
    }
    float m = fmaxf(s0, fmaxf(s1, s2));
    float e0 = __expf(s0 - m), e1 = __expf(s1 - m), e2 = __expf(s2 - m);
    float inv = 1.0f / (e0 + e1 + e2);
    float w0 = e0 * inv, w1 = e1 * inv, w2 = e2 * inv;
    for (int j = lane; j < H; j += 32)
        fused[base + j] = w0 * kg[base + j] + w1 * im[base + j] + w2 * tx[base + j];
}

// ---------------------------------------------------------------------------
// fp32 WMMA GEMM: C[M,N] = A[M,K] @ B[K,N] (+bias), epilogue modes:
//   0: plain   1: ReLU   2: +resid   3: pp2 permuted scatter
// One wave computes a 16x64 tile (4 accumulators) via V_WMMA_F32_16X16X4_F32:
// one A fragment (b64 load) feeds 4 WMMAs -> 4x A reuse, better WMMA:VMEM mix.
// Requires M % 16 == 0 and N % 64 == 0 (true for all shapes here).
// ---------------------------------------------------------------------------
__global__ void wmma_gemm_f32(const float* __restrict__ A, const float* __restrict__ B,
                              const float* __restrict__ bias, const float* __restrict__ resid,
                              float* __restrict__ C, int M, int N, int K, int mode) {
    int wave = blockIdx.x * (blockDim.x >> 5) + (threadIdx.x >> 5);
    int ntN = N >> 6;                         // 64-wide N tiles
    int ntM = M >> 4;
    if (wave >= ntM * ntN) return;            // wave-uniform: EXEC stays all-ones
    int tm = wave / ntN;
    int tn = wave - tm * ntN;
    int lane = threadIdx.x & 31;
    int g = lane >> 4;                        // half-wave group
    int ln = lane & 15;
    const float* Ap = A + (size_t)(tm * 16 + ln) * K;
    const float* Bp = B + (tn * 64 + ln);
    int kg2 = 2 * g;
    v8f acc0 = {}, acc1 = {}, acc2 = {}, acc3 = {};
#pragma unroll 2
    for (int k = 0; k < K; k += 4) {
        int kk = k + kg2;
        v2f a = *(const v2f*)(Ap + kk);       // kk even, 8B-aligned -> b64 load
        const float* b0 = Bp + (size_t)kk * N;
        const float* b1 = Bp + (size_t)(kk + 1) * N;
        v2f b;
        b.x = b0[0];  b.y = b1[0];
        acc0 = __builtin_amdgcn_wmma_f32_16x16x4_f32(false, a, false, b, (short)0, acc0,
                                                     false, false);
        b.x = b0[16]; b.y = b1[16];
        acc1 = __builtin_amdgcn_wmma_f32_16x16x4_f32(false, a, false, b, (short)0, acc1,
                                                     false, false);
        b.x = b0[32]; b.y = b1[32];
        acc2 = __builtin_amdgcn_wmma_f32_16x16x4_f32(false, a, false, b, (short)0, acc2,
                                                     false, false);
        b.x = b0[48]; b.y = b1[48];
        acc3 = __builtin_amdgcn_wmma_f32_16x16x4_f32(false, a, false, b, (short)0, acc3,
                                                     false, false);
    }
#pragma unroll
    for (int t = 0; t < 4; ++t) {
        v8f acc = (t == 0) ? acc0 : (t == 1) ? acc1 : (t == 2) ? acc2 : acc3;
        int col = tn * 64 + t * 16 + ln;
        float bv = bias ? bias[col] : 0.0f;
#pragma unroll
        for (int v = 0; v < 8; ++v) {
            int row = tm * 16 + v + 8 * g;
            float val = acc[v] + bv;
            if (mode == 1) val = fmaxf(val, 0.0f);
            if (mode == 2) val += resid[(size_t)row * N + col];
            if (mode == 3) {
                // col -> (layer, block, head, hd); row -> (b, l)
                int layer = col >> 10;
                int rem   = col & 1023;
                int blk   = rem >> 9;
                int rem2  = rem & 511;
                int head  = rem2 >> 6;
                int hd    = rem2 & 63;
                int bb    = row / ELEN;
                int l     = row - bb * ELEN;
                size_t oidx = ((((size_t)(layer * NBLOCK + blk) * BATCH + bb) * NHEAD + head)
                               * ELEN + l) * (H / NHEAD) + hd;
                C[oidx] = val;
            } else {
                C[(size_t)row * N + col] = val;
            }
        }
    }
}

__global__ void write_tail(float* out, size_t idx) {
    if (blockIdx.x == 0 && threadIdx.x == 0) out[idx] = 0.0f;
}

// ---------------------------------------------------------------------------
// Host orchestration
// ---------------------------------------------------------------------------
static inline int cdiv(long long a, long long b) { return (int)((a + b - 1) / b); }

extern "C" void kernel_launch(void* const* d_in, const int* in_sizes, int n_in,
                              void* d_out, int out_size, void* d_ws, size_t ws_size,
                              hipStream_t stream) {
    // ---- inputs (setup_inputs order) ----
    const int*   entity_ids   = (const int*)  d_in[0];
    const float* token_embeds = (const float*)d_in[1];
    const int*   edge_index   = (const int*)  d_in[2];
    const int*   edge_type    = (const int*)  d_in[3];
    const float* node_embeds  = (const float*)d_in[4];
    const float* rgcn_basis   = (const float*)d_in[5];
    const float* rgcn_comp    = (const float*)d_in[6];
    const float* rgcn_root    = (const float*)d_in[7];
    const float* rgcn_bias    = (const float*)d_in[8];
    const float* kg_proj_w    = (const float*)d_in[9];
    const float* kg_proj_b    = (const float*)d_in[10];
    const float* text_proj_w  = (const float*)d_in[11];
    const float* text_proj_b  = (const float*)d_in[12];
    const float* image_proj_w = (const float*)d_in[13];
    const float* image_proj_b = (const float*)d_in[14];
    const float* cq_w         = (const float*)d_in[15];
    const float* cq_b         = (const float*)d_in[16];
    const float* kgk_w        = (const float*)d_in[17];
    const float* kgk_b        = (const float*)d_in[18];
    const float* imk_w        = (const float*)d_in[19];
    const float* imk_b        = (const float*)d_in[20];
    const float* txk_w        = (const float*)d_in[21];
    const float* txk_b        = (const float*)d_in[22];
    const float* pp1_w1       = (const float*)d_in[23];
    const float* pp1_b1       = (const float*)d_in[24];
    const float* pp1_w2       = (const float*)d_in[25];
    const float* pp1_b2       = (const float*)d_in[26];
    const float* pp2_w        = (const float*)d_in[27];
    const float* pp2_b        = (const float*)d_in[28];
    const float* text_init    = (const float*)d_in[29];
    const float* image_init   = (const float*)d_in[30];
    float* out = (float*)d_out;

    // ---- workspace carve-out (256B aligned) ----
    char* wsp = (char*)d_ws;
    auto carve = [&](size_t bytes) -> void* {
        void* p = (void*)wsp;
        wsp += (bytes + 255) & ~(size_t)255;
        return p;
    };
    int*   slot_map     = (int*)  carve((size_t)NENT * 4);
    int*   counter      = (int*)  carve(4);
    int*   node_of_slot = (int*)  carve((size_t)NROW * 4);
    float* cnt_c        = (float*)carve((size_t)NROW * NREL * 4);
    float* agg_c        = (float*)carve((size_t)NROW * NREL * HALF * 4);
    float* wcat         = (float*)carve((size_t)KCAT * HALF * 4);
    float* acat         = (float*)carve((size_t)NROW * KCAT * 4);
    float* rgcn_out     = (float*)carve((size_t)NROW * HALF * 4);
    float* kg_allc      = (float*)carve((size_t)NROW * H * 4);
    float* kg_feat      = (float*)carve((size_t)NROW * H * 4);
    float* tg           = (float*)carve((size_t)NROW * TXTD * 4);
    float* ig           = (float*)carve((size_t)NROW * TXTD * 4);
    float* tfeat        = (float*)carve((size_t)NROW * H * 4);
    float* ifeat        = (float*)carve((size_t)NROW * H * 4);
    float* qin          = (float*)carve((size_t)BATCH * H * 4);
    float* qbuf         = (float*)carve((size_t)BATCH * H * 4);
    float* kkb          = (float*)carve((size_t)NROW * H * 4);
    float* ikb          = (float*)carve((size_t)NROW * H * 4);
    float* tkb          = (float*)carve((size_t)NROW * H * 4);
    float* fused        = (float*)carve((size_t)NROW * H * 4);
    float* h1           = (float*)carve((size_t)NROW * HALF * 4);
    float* hbuf         = (float*)carve((size_t)NROW * H * 4);
    (void)ws_size; (void)n_in; (void)in_sizes;

    const int T = 256;
    // ---- init scratch (fresh every call; ws is not re-poisoned between replays) ----
    fill_i32<<<cdiv(NENT, T), T, 0, stream>>>(slot_map, -1, NENT);
    fill_i32<<<1, T, 0, stream>>>(counter, 0, 1);
    fill_i32<<<cdiv(NROW, T), T, 0, stream>>>(node_of_slot, 0, NROW);
    fill_f32<<<cdiv(NROW * NREL, T), T, 0, stream>>>(cnt_c, 0.f, NROW * NREL);
    fill_f32<<<cdiv(NROW * NREL * HALF, T), T, 0, stream>>>(agg_c, 0.f, NROW * NREL * HALF);

    // ---- compact slots + RGCN pipeline ----
    assign_slots<<<cdiv(NROW, T), T, 0, stream>>>(entity_ids, slot_map, counter, node_of_slot);
    build_wcat<<<cdiv(KCAT * HALF, T), T, 0, stream>>>(rgcn_basis, rgcn_comp, rgcn_root, wcat);
    edge_scatter<<<cdiv(NEDGE, T / 32), T, 0, stream>>>(edge_index, edge_type, slot_map,
                                                        node_embeds, agg_c, cnt_c);
    build_acat<<<cdiv(NROW * KCAT, T), T, 0, stream>>>(agg_c, cnt_c, node_of_slot,
                                                       node_embeds, acat);

    auto gemm = [&](const float* A, const float* B, const float* bias, const float* resid,
                    float* C, int M, int N, int K, int mode) {
        int tiles = (M / 16) * (N / 64);
        wmma_gemm_f32<<<cdiv(tiles, T / 32), T, 0, stream>>>(A, B, bias, resid, C, M, N, K, mode);
    };

    // RGCN: [1600,3328] @ [3328,256] + bias  (mean-msgs + root fused)
    gemm(acat, wcat, rgcn_bias, nullptr, rgcn_out, NROW, HALF, KCAT, 0);
    // kg projection: [1600,256] @ [256,512] + b
    gemm(rgcn_out, kg_proj_w, kg_proj_b, nullptr, kg_allc, NROW, H, HALF, 0);

    // ---- gathers ----
    gather_rows<<<cdiv(NROW * H, T), T, 0, stream>>>(kg_allc, entity_ids, slot_map, kg_feat, H);
    gather_rows<<<cdiv(NROW * TXTD, T), T, 0, stream>>>(text_init, entity_ids, nullptr, tg, TXTD);
    gather_rows<<<cdiv(NROW * TXTD, T), T, 0, stream>>>(image_init, entity_ids, nullptr, ig, TXTD);

    // ---- modality projections ----
    gemm(tg, text_proj_w, text_proj_b, nullptr, tfeat, NROW, H, TXTD, 0);
    gemm(ig, image_proj_w, image_proj_b, nullptr, ifeat, NROW, H, TXTD, 0);

    // ---- context query ----
    qmean<<<cdiv(BATCH * H, T), T, 0, stream>>>(token_embeds, qin);
    gemm(qin, cq_w, cq_b, nullptr, qbuf, BATCH, H, H, 0);

    // ---- key projections + gated fusion ----
    gemm(kg_feat, kgk_w, kgk_b, nullptr, kkb, NROW, H, H, 0);
    gemm(ifeat, imk_w, imk_b, nullptr, ikb, NROW, H, H, 0);
    gemm(tfeat, txk_w, txk_b, nullptr, tkb, NROW, H, H, 0);
    fuse_kernel<<<cdiv(NROW, T / 32), T, 0, stream>>>(qbuf, kg_feat, kkb, ifeat, ikb,
                                                      tfeat, tkb, fused);

    // ---- prompt MLP with residual ----
    gemm(fused, pp1_w1, pp1_b1, nullptr, h1, NROW, HALF, H, 1);   // ReLU
    gemm(h1, pp1_w2, pp1_b2, fused, hbuf, NROW, H, HALF, 2);      // +fused residual

    // ---- pp2 with fused permuted scatter into d_out ----
    gemm(hbuf, pp2_w, pp2_b, nullptr, out, NROW, PPN, H, 3);

    // ---- loss_cl = 0.0 appended after p ----
    write_tail<<<1, 32, 0, stream>>>(out, (size_t)out_size - 1);
}